// GAT_21337397527228
// MI455X (gfx1250) — compile-verified
//
#include <hip/hip_runtime.h>
#include <hip/hip_bf16.h>
#include <math.h>

typedef __attribute__((ext_vector_type(2))) float v2f;
typedef __attribute__((ext_vector_type(8))) float v8f;

#define NEG_SLOPE 0.2f

// ---------------------------------------------------------------------------
// C[M x 128] = A[M x 128] @ W[128 x 128]  (+ bias0[col] + bias1[col])
// block = 256 threads (8 wave32) covering a 128x128 output tile.
// Wave grid 4x2: wave w -> rows [blk*128 + (w>>1)*32, +32), cols [(w&1)*64, +64)
//   -> per k-step: 2 A-frag loads + 4 B-frag loads feed 8 v_wmma_f32_16x16x4_f32.
// W staged in LDS with paired-K layout lwp[(k>>1)*256 + n*2 + (k&1)] so a
// B-fragment is one aligned ds_load_b64 straight into an even VGPR pair.
// ---------------------------------------------------------------------------
__global__ __launch_bounds__(256) void gat_gemm128(
    const float* __restrict__ A, const float* __restrict__ W,
    const float* __restrict__ bias0, const float* __restrict__ bias1,
    float* __restrict__ C, int M)
{
    __shared__ float lwp[128 * 128];    // 64 KB, paired-K layout
    const int t = threadIdx.x;
    const int wave = t >> 5;
    const int lane = t & 31;
    const int half = lane >> 4;         // lane group 0/1
    const int l16  = lane & 15;
    const int row0 = blockIdx.x * 128 + (wave >> 1) * 32;
    const int col0 = (wave & 1) * 64;

    const int r0c = row0 + l16;
    const int r1c = row0 + 16 + l16;
    const float* Ar0 = A + (size_t)((r0c < M) ? r0c : (M - 1)) * 128;
    const float* Ar1 = A + (size_t)((r1c < M) ? r1c : (M - 1)) * 128;
    // pull this wave's A rows toward the caches while we stage W + barrier
    __builtin_prefetch(Ar0, 0, 1);
    __builtin_prefetch(Ar0 + 64, 0, 1);
    __builtin_prefetch(Ar1, 0, 1);
    __builtin_prefetch(Ar1 + 64, 0, 1);

    // stage W: read two rows (k = 2q, 2q+1) coalesced, write K-interleaved pairs
    for (int p = t; p < 64 * 32; p += 256) {        // 8 iterations
        const int q  = p >> 5;                      // k-pair index 0..63
        const int n4 = (p & 31) << 2;               // column 0..124 step 4
        const float4 ra = *(const float4*)(W + (2 * q) * 128 + n4);
        const float4 rb = *(const float4*)(W + (2 * q + 1) * 128 + n4);
        float4* dst = (float4*)(lwp + q * 256 + n4 * 2);
        dst[0] = make_float4(ra.x, rb.x, ra.y, rb.y);
        dst[1] = make_float4(ra.z, rb.z, ra.w, rb.w);
    }
    __syncthreads();

    // accumulators: [row tile 0/1][col tile 0..3], init with column bias
    v8f acc[2][4];
    #pragma unroll
    for (int cb = 0; cb < 4; ++cb) {
        float bv = 0.0f;
        const int n = col0 + cb * 16 + l16;
        if (bias0) bv += bias0[n];
        if (bias1) bv += bias1[n];
        #pragma unroll
        for (int r = 0; r < 8; ++r) { acc[0][cb][r] = bv; acc[1][cb][r] = bv; }
    }

    #pragma unroll 4
    for (int k0 = 0; k0 < 128; k0 += 4) {
        const int ka = k0 + 2 * half;               // even
        const v2f a0 = *(const v2f*)(Ar0 + ka);     // A[m, ka], A[m, ka+1]
        const v2f a1 = *(const v2f*)(Ar1 + ka);
        const float* lb = lwp + (ka >> 1) * 256 + col0 * 2 + l16 * 2;
        #pragma unroll
        for (int cb = 0; cb < 4; ++cb) {
            const v2f b = *(const v2f*)(lb + cb * 32);   // {W[ka,n], W[ka+1,n]}
            acc[0][cb] = __builtin_amdgcn_wmma_f32_16x16x4_f32(
                false, a0, false, b, (short)0, acc[0][cb], false, false);
            acc[1][cb] = __builtin_amdgcn_wmma_f32_16x16x4_f32(
                false, a1, false, b, (short)0, acc[1][cb], false, false);
        }
    }

    // D layout: vgpr r -> row (r + 8*half), col = col0 + cb*16 + l16
    #pragma unroll
    for (int tr = 0; tr < 2; ++tr) {
        #pragma unroll
        for (int r = 0; r < 8; ++r) {
            const int m = row0 + tr * 16 + r + 8 * half;
            if (m < M) {
                #pragma unroll
                for (int cb = 0; cb < 4; ++cb)
                    C[(size_t)m * 128 + col0 + cb * 16 + l16] = acc[tr][cb][r];
            }
        }
    }
}

// ---------------------------------------------------------------------------
// out[i] = dot(H[i, 0:128], v[0:128]) ; one wave32 per row, float4 per lane
// ---------------------------------------------------------------------------
__global__ __launch_bounds__(256) void gat_rowdot(
    const float* __restrict__ H, const float* __restrict__ v,
    float* __restrict__ out, int rows)
{
    const int gw = (blockIdx.x * 256 + threadIdx.x) >> 5;
    const int lane = threadIdx.x & 31;
    if (gw >= rows) return;
    const float4 h4 = ((const float4*)(H + (size_t)gw * 128))[lane];
    const float4 v4 = ((const float4*)v)[lane];
    float s = h4.x * v4.x + h4.y * v4.y + h4.z * v4.z + h4.w * v4.w;
    #pragma unroll
    for (int off = 16; off > 0; off >>= 1)
        s += __shfl_down(s, off, 32);
    if (lane == 0) out[gw] = s;
}

__global__ void gat_init(float* __restrict__ emax, float* __restrict__ esum, int n)
{
    const int i = blockIdx.x * blockDim.x + threadIdx.x;
    if (i < n) { emax[i] = -INFINITY; esum[i] = 0.0f; }
}

__device__ __forceinline__ float leaky(float v)
{
    return (v > 0.0f) ? v : NEG_SLOPE * v;
}

__global__ void gat_edge_max(const int* __restrict__ src, const int* __restrict__ dst,
                             const float* __restrict__ as, const float* __restrict__ ad,
                             float* __restrict__ emax, int E)
{
    const int e = blockIdx.x * blockDim.x + threadIdx.x;
    if (e >= E) return;
    const int s = src[e], d = dst[e];
    atomicMax(emax + d, leaky(as[s] + ad[d]));
}

__global__ void gat_edge_sum(const int* __restrict__ src, const int* __restrict__ dst,
                             const float* __restrict__ as, const float* __restrict__ ad,
                             const float* __restrict__ emax, float* __restrict__ esum, int E)
{
    const int e = blockIdx.x * blockDim.x + threadIdx.x;
    if (e >= E) return;
    const int s = src[e], d = dst[e];
    atomicAdd(esum + d, __expf(leaky(as[s] + ad[d]) - emax[d]));
}

// one wave32 per edge: out[dst] += alpha * hsrc[src]; lane covers 4 columns
__global__ __launch_bounds__(256) void gat_edge_agg(
    const int* __restrict__ src, const int* __restrict__ dst,
    const float* __restrict__ as, const float* __restrict__ ad,
    const float* __restrict__ emax, const float* __restrict__ esum,
    const float* __restrict__ hsrc, float* __restrict__ outacc, int E)
{
    const int gw = (blockIdx.x * 256 + threadIdx.x) >> 5;
    const int lane = threadIdx.x & 31;
    if (gw >= E) return;
    const int s = src[gw], d = dst[gw];
    const float alpha = __expf(leaky(as[s] + ad[d]) - emax[d]) / (esum[d] + 1e-16f);
    const float4 m = ((const float4*)(hsrc + (size_t)s * 128))[lane];
    float* o = outacc + (size_t)d * 128 + lane * 4;
    atomicAdd(o + 0, m.x * alpha);
    atomicAdd(o + 1, m.y * alpha);
    atomicAdd(o + 2, m.z * alpha);
    atomicAdd(o + 3, m.w * alpha);
}

__global__ void gat_relu(float* __restrict__ a, int n)
{
    const int i = blockIdx.x * blockDim.x + threadIdx.x;
    if (i < n) a[i] = fmaxf(a[i], 0.0f);
}

// ---------------------------------------------------------------------------
extern "C" void kernel_launch(void* const* d_in, const int* in_sizes, int n_in,
                              void* d_out, int out_size, void* d_ws, size_t ws_size,
                              hipStream_t stream)
{
    const float* x   = (const float*)d_in[0];
    const int*   ei  = (const int*)  d_in[1];
    const float* W1s = (const float*)d_in[2];
    const float* W1d = (const float*)d_in[3];
    const float* a1s = (const float*)d_in[4];
    const float* a1d = (const float*)d_in[5];
    const float* b1  = (const float*)d_in[6];
    const float* Wl1 = (const float*)d_in[7];
    const float* bl1 = (const float*)d_in[8];
    const float* W2s = (const float*)d_in[9];
    const float* W2d = (const float*)d_in[10];
    const float* a2s = (const float*)d_in[11];
    const float* a2d = (const float*)d_in[12];
    const float* b2  = (const float*)d_in[13];
    const float* Wl2 = (const float*)d_in[14];
    const float* bl2 = (const float*)d_in[15];
    float* out = (float*)d_out;

    const int N = in_sizes[0] / 128;
    const int E = in_sizes[1] / 2;
    const int* src = ei;
    const int* dst = ei + E;

    // workspace layout (f32)
    float* hsrc = (float*)d_ws;                 // N*128
    float* hbuf = hsrc + (size_t)N * 128;       // N*128 : skip-accum -> h
    float* asrc = hbuf + (size_t)N * 128;       // N
    float* adst = asrc + N;                     // N
    float* emax = adst + N;                     // N
    float* esum = emax + N;                     // N
    float* vtmp = esum + N;                     // 128

    const dim3 blk(256);
    const int gGemm  = (N + 127) / 128;
    const int gWave  = (N * 32 + 255) / 256;
    const int gNode  = (N + 255) / 256;
    const int gEdge  = (E + 255) / 256;
    const int gEdgeW = (E + 7) / 8;             // one wave per edge
    const int gElem  = (N * 128 + 255) / 256;

    // ------------------------- layer 1 -------------------------
    gat_gemm128<<<gGemm, blk, 0, stream>>>(x, W1s, nullptr, nullptr, hsrc, N);
    gat_gemm128<<<gGemm, blk, 0, stream>>>(x, Wl1, bl1, b1, hbuf, N);   // skip + both biases
    gat_rowdot<<<16, blk, 0, stream>>>(W1d, a1d, vtmp, 128);            // vtmp = W1d @ a1d
    gat_rowdot<<<gWave, blk, 0, stream>>>(x, vtmp, adst, N);            // alpha_dst = x @ vtmp
    gat_rowdot<<<gWave, blk, 0, stream>>>(hsrc, a1s, asrc, N);          // alpha_src
    gat_init<<<gNode, blk, 0, stream>>>(emax, esum, N);
    gat_edge_max<<<gEdge, blk, 0, stream>>>(src, dst, asrc, adst, emax, E);
    gat_edge_sum<<<gEdge, blk, 0, stream>>>(src, dst, asrc, adst, emax, esum, E);
    gat_edge_agg<<<gEdgeW, blk, 0, stream>>>(src, dst, asrc, adst, emax, esum, hsrc, hbuf, E);
    gat_relu<<<gElem, blk, 0, stream>>>(hbuf, N * 128);                 // h = relu(...)

    // ------------------------- layer 2 -------------------------
    gat_gemm128<<<gGemm, blk, 0, stream>>>(hbuf, W2s, nullptr, nullptr, hsrc, N);
    gat_gemm128<<<gGemm, blk, 0, stream>>>(hbuf, Wl2, bl2, b2, out, N); // accum in d_out
    gat_rowdot<<<16, blk, 0, stream>>>(W2d, a2d, vtmp, 128);
    gat_rowdot<<<gWave, blk, 0, stream>>>(hbuf, vtmp, adst, N);
    gat_rowdot<<<gWave, blk, 0, stream>>>(hsrc, a2s, asrc, N);
    gat_init<<<gNode, blk, 0, stream>>>(emax, esum, N);
    gat_edge_max<<<gEdge, blk, 0, stream>>>(src, dst, asrc, adst, emax, E);
    gat_edge_sum<<<gEdge, blk, 0, stream>>>(src, dst, asrc, adst, emax, esum, E);
    gat_edge_agg<<<gEdgeW, blk, 0, stream>>>(src, dst, asrc, adst, emax, esum, hsrc, out, E);
}